// TreeLSTM_90177133347396
// MI455X (gfx1250) — compile-verified
//
#include <hip/hip_runtime.h>
#include <hip/hip_bf16.h>

#define DEPTH 17
#define NTOT ((1 << DEPTH) - 1)   // 131071
#define NFEAT 300
#define NHID 256
#define NIOU (3 * NHID)           // 768
#define NCLASS 100

typedef __attribute__((ext_vector_type(2))) float v2f;
typedef __attribute__((ext_vector_type(8))) float v8f;

__device__ __forceinline__ float sigmoidf_(float x) {
    return 1.0f / (1.0f + __expf(-x));
}

// ---------------------------------------------------------------------------
// C[M,NCOL] = A[M,K] @ W[K,NCOL] + bias   (f32 WMMA 16x16x4)
// One 16-row tile per workgroup; A tile staged in LDS; 8 waves, one 16-col
// tile each; grid.y covers NCOL/128.
// ---------------------------------------------------------------------------
template<int KDIM, int NCOL>
__global__ __launch_bounds__(256)
void gemm_bias_wmma(const float* __restrict__ A, const float* __restrict__ W,
                    const float* __restrict__ bias, float* __restrict__ C, int M) {
    constexpr int KPAD = KDIM + 4;
    __shared__ float lds_a[16 * KPAD];

    const int tid  = threadIdx.x;
    const int row0 = blockIdx.x * 16;

    for (int idx = tid; idx < 16 * KDIM; idx += 256) {
        int r = idx / KDIM, k = idx - r * KDIM;
        float v = 0.0f;
        int grow = row0 + r;
        if (grow < M) v = A[(long)grow * KDIM + k];
        lds_a[r * KPAD + k] = v;
    }
    __syncthreads();

    const int wave = tid >> 5;
    const int lane = tid & 31;
    const int half = lane >> 4;          // K sub-half per ISA A-layout
    const int m16  = lane & 15;
    const int col  = (blockIdx.y * 8 + wave) * 16 + m16;

    v8f acc = {};
    for (int k0 = 0; k0 < KDIM; k0 += 4) {
        int ka = k0 + 2 * half;
        v2f a = *(const v2f*)&lds_a[m16 * KPAD + ka];
        v2f b;
        b.x = W[(long)ka * NCOL + col];
        b.y = W[(long)(ka + 1) * NCOL + col];
        acc = __builtin_amdgcn_wmma_f32_16x16x4_f32(false, a, false, b,
                                                    (short)0, acc, false, false);
    }
    float bcol = bias[col];
#pragma unroll
    for (int v = 0; v < 8; ++v) {
        int grow = row0 + v + 8 * half;
        if (grow < M) C[(long)grow * NCOL + col] = acc[v] + bcol;
    }
}

// ---------------------------------------------------------------------------
// Leaf level (d = DEPTH-1): children are zero -> no GEMMs needed.
// iou = iou_x + b_iouh ; c = sig(i)*tanh(u) ; h = sig(o)*tanh(c)
// ---------------------------------------------------------------------------
__global__ __launch_bounds__(256)
void leaf_kernel(const float* __restrict__ iou_x, const float* __restrict__ b_iouh,
                 float* __restrict__ h, float* __restrict__ c, int s0, int n) {
    long total = (long)n * NHID;
    for (long t = (long)blockIdx.x * blockDim.x + threadIdx.x; t < total;
         t += (long)gridDim.x * blockDim.x) {
        int  j    = (int)(t >> 8);
        int  col  = (int)(t & 255);
        long node = (long)s0 + j;
        const float* iou = iou_x + node * NIOU;
        float ig = sigmoidf_(iou[col]            + b_iouh[col]);
        float og = sigmoidf_(iou[NHID + col]     + b_iouh[NHID + col]);
        float ug = tanhf    (iou[2 * NHID + col] + b_iouh[2 * NHID + col]);
        float cn = ig * ug;
        c[node * NHID + col] = cn;
        h[node * NHID + col] = og * tanhf(cn);
    }
}

// ---------------------------------------------------------------------------
// Internal level: one 16-node tile per workgroup, 512 threads = 16 waves.
// Wave w owns hidden block w (16 cols) and accumulates the 5 output tiles
// (i, o, u at cols {w,w+16,w+32} of W_iouh ; f0,f1 at col w of W_fh) over a
// single K=256 loop, then fuses the gate math entirely in registers.
// ---------------------------------------------------------------------------
__global__ __launch_bounds__(512)
void level_kernel(const float* __restrict__ iou_x, const float* __restrict__ fx,
                  const float* __restrict__ W_iouh, const float* __restrict__ b_iouh,
                  const float* __restrict__ W_fh,   const float* __restrict__ b_fh,
                  float* __restrict__ h, float* __restrict__ c,
                  int s0, int n) {
    __shared__ float lds_hsum[16 * NHID];
    __shared__ float lds_h0[16 * NHID];
    __shared__ float lds_h1[16 * NHID];

    const int tid = threadIdx.x;
    const int j0  = blockIdx.x * 16;
    const int cs  = 2 * s0 + 1;

    for (int idx = tid; idx < 16 * NHID; idx += 512) {
        int m = idx >> 8, colh = idx & 255;
        int j = j0 + m;
        float h0v = 0.0f, h1v = 0.0f;
        if (j < n) {
            long cbase = (long)(cs + 2 * j) * NHID + colh;
            h0v = h[cbase];
            h1v = h[cbase + NHID];
        }
        lds_h0[idx]   = h0v;
        lds_h1[idx]   = h1v;
        lds_hsum[idx] = h0v + h1v;
    }
    __syncthreads();

    const int wave = tid >> 5;            // hidden block 0..15
    const int lane = tid & 31;
    const int half = lane >> 4;
    const int m16  = lane & 15;
    const int colg = wave * 16 + m16;     // hidden column 0..255

    v8f acc_i = {}, acc_o = {}, acc_u = {}, acc_f0 = {}, acc_f1 = {};
    for (int k0 = 0; k0 < NHID; k0 += 4) {
        int ka = k0 + 2 * half;
        v2f a_hs = *(const v2f*)&lds_hsum[m16 * NHID + ka];
        v2f a_h0 = *(const v2f*)&lds_h0[m16 * NHID + ka];
        v2f a_h1 = *(const v2f*)&lds_h1[m16 * NHID + ka];
        const float* wr = W_iouh + (long)ka * NIOU;
        const float* wf = W_fh   + (long)ka * NHID;
        v2f b_i, b_o, b_u, b_f;
        b_i.x = wr[colg];            b_i.y = wr[NIOU + colg];
        b_o.x = wr[NHID + colg];     b_o.y = wr[NIOU + NHID + colg];
        b_u.x = wr[2 * NHID + colg]; b_u.y = wr[NIOU + 2 * NHID + colg];
        b_f.x = wf[colg];            b_f.y = wf[NHID + colg];
        acc_i  = __builtin_amdgcn_wmma_f32_16x16x4_f32(false, a_hs, false, b_i, (short)0, acc_i,  false, false);
        acc_o  = __builtin_amdgcn_wmma_f32_16x16x4_f32(false, a_hs, false, b_o, (short)0, acc_o,  false, false);
        acc_u  = __builtin_amdgcn_wmma_f32_16x16x4_f32(false, a_hs, false, b_u, (short)0, acc_u,  false, false);
        acc_f0 = __builtin_amdgcn_wmma_f32_16x16x4_f32(false, a_h0, false, b_f, (short)0, acc_f0, false, false);
        acc_f1 = __builtin_amdgcn_wmma_f32_16x16x4_f32(false, a_h1, false, b_f, (short)0, acc_f1, false, false);
    }

    const float b_ic = b_iouh[colg];
    const float b_oc = b_iouh[NHID + colg];
    const float b_uc = b_iouh[2 * NHID + colg];
    const float b_fc = b_fh[colg];
#pragma unroll
    for (int v = 0; v < 8; ++v) {
        int m = v + 8 * half;             // C-matrix row per ISA layout
        int j = j0 + m;
        if (j < n) {
            long node = (long)s0 + j;
            const float* iou = iou_x + node * NIOU;
            float ig  = sigmoidf_(acc_i[v]  + iou[colg]            + b_ic);
            float og  = sigmoidf_(acc_o[v]  + iou[NHID + colg]     + b_oc);
            float ug  = tanhf    (acc_u[v]  + iou[2 * NHID + colg] + b_uc);
            float fxv = fx[node * NHID + colg] + b_fc;
            float f0  = sigmoidf_(acc_f0[v] + fxv);
            float f1  = sigmoidf_(acc_f1[v] + fxv);
            long cbase = (long)(cs + 2 * j) * NHID + colg;
            float cn = ig * ug + f0 * c[cbase] + f1 * c[cbase + NHID];
            c[node * NHID + colg] = cn;
            h[node * NHID + colg] = og * tanhf(cn);
        }
    }
}

// ---------------------------------------------------------------------------
// out[0:100] = relu(h[node 0]) @ W_ln + b_ln
// ---------------------------------------------------------------------------
__global__ __launch_bounds__(128)
void root_kernel(const float* __restrict__ h, const float* __restrict__ W_ln,
                 const float* __restrict__ b_ln, float* __restrict__ out) {
    __shared__ float xs[NHID];
    for (int k = threadIdx.x; k < NHID; k += 128) xs[k] = fmaxf(h[k], 0.0f);
    __syncthreads();
    int t = threadIdx.x;
    if (t < NCLASS) {
        float acc = b_ln[t];
        for (int k = 0; k < NHID; ++k) acc = fmaf(xs[k], W_ln[k * NCLASS + t], acc);
        out[t] = acc;
    }
}

extern "C" void kernel_launch(void* const* d_in, const int* in_sizes, int n_in,
                              void* d_out, int out_size, void* d_ws, size_t ws_size,
                              hipStream_t stream) {
    const float* features = (const float*)d_in[0];
    const float* W_ioux   = (const float*)d_in[1];
    const float* b_ioux   = (const float*)d_in[2];
    const float* W_iouh   = (const float*)d_in[3];
    const float* b_iouh   = (const float*)d_in[4];
    const float* W_fx     = (const float*)d_in[5];
    const float* b_fx     = (const float*)d_in[6];
    const float* W_fh     = (const float*)d_in[7];
    const float* b_fh     = (const float*)d_in[8];
    const float* W_ln     = (const float*)d_in[9];
    const float* b_ln     = (const float*)d_in[10];
    float* out = (float*)d_out;

    // Workspace layout (floats): iou_x | fx | h | c
    float* iou_x = (float*)d_ws;
    float* fx    = iou_x + (long)NTOT * NIOU;
    float* h     = fx    + (long)NTOT * NHID;
    float* c     = h     + (long)NTOT * NHID;

    const int mtiles = (NTOT + 15) / 16;   // 8192

    // Precompute input projections (f32 WMMA GEMMs)
    gemm_bias_wmma<NFEAT, NIOU><<<dim3(mtiles, NIOU / 128), 256, 0, stream>>>(
        features, W_ioux, b_ioux, iou_x, NTOT);
    gemm_bias_wmma<NFEAT, NHID><<<dim3(mtiles, NHID / 128), 256, 0, stream>>>(
        features, W_fx, b_fx, fx, NTOT);

    // Leaf level (d = 16): children are zero, pure elementwise
    {
        int s0 = (1 << (DEPTH - 1)) - 1;   // 65535
        int n  = 1 << (DEPTH - 1);         // 65536
        leaf_kernel<<<16384, 256, 0, stream>>>(iou_x, b_iouh, h, c, s0, n);
    }

    // Internal levels d = 15 .. 0 (sequentially dependent)
    for (int d = DEPTH - 2; d >= 0; --d) {
        int s0 = (1 << d) - 1;
        int n  = 1 << d;
        int blocks = (n + 15) / 16;
        level_kernel<<<blocks, 512, 0, stream>>>(iou_x, fx, W_iouh, b_iouh,
                                                 W_fh, b_fh, h, c, s0, n);
    }

    // Head: relu(h_root) @ W_ln + b_ln
    root_kernel<<<1, 128, 0, stream>>>(h, W_ln, b_ln, out);
}